// ContinuousEmbedding_15032385536233
// MI455X (gfx1250) — compile-verified
//
#include <hip/hip_runtime.h>
#include <cstdint>

// ContinuousEmbedding (Hann-window interpolation into a 2048x64 table).
// Memory/latency-bound gather op: ~26 MFLOP vs ~7 MB of traffic.
// CDNA5 path used: per-wave async DMA (global_load_async_to_lds_b128 ->
// ASYNCcnt -> s_wait_asynccnt) to stage each x-value's 8x64 f32 embedding
// window (2 KB) into LDS, overlapped with the v_cos_f32 weight computation;
// consumption via coalesced ds_load_b64. WMMA is intentionally NOT used:
// every output row gathers a different 8-row window, so no B-fragment can be
// shared across a 16-row WMMA tile; dense-K WMMA would waste 256x FLOPs on a
// bandwidth-bound kernel.

#define CE_NUM_POINTS 2048
#define CE_OUT_DIMS   64
#define CE_WAVES_PER_BLOCK 8
#define CE_SLOT_FLOATS (8 * CE_OUT_DIMS)   // 8 rows * 64 f32 = 2 KB per wave

__global__ __launch_bounds__(256)
void ContinuousEmbedding_kernel(const float* __restrict__ x,
                                const float* __restrict__ emb,
                                float* __restrict__ out,
                                int nx)
{
    __shared__ __align__(16) float lds[CE_WAVES_PER_BLOCK * CE_SLOT_FLOATS]; // 16 KB

    const int lane = threadIdx.x & 31;
    const int wave = threadIdx.x >> 5;
    const int xi   = blockIdx.x * CE_WAVES_PER_BLOCK + wave;
    if (xi >= nx) return;   // per-wave uniform branch; no block barriers used

    // xs in point-index space: (x - MINVAL) * NUM_POINTS / (MAXVAL - MINVAL)
    const float xv = x[xi];
    const float xs = (xv + 1.0f) * (0.5f * (float)CE_NUM_POINTS);
    const int   p0 = (int)floorf(xs) - 3;        // first of 8 window taps

    // Clamp the staged window so the DMA stays in-bounds; taps outside
    // [0, NUM_POINTS) get zero weight below, matching the reference.
    int pstage = p0;
    pstage = pstage < 0 ? 0 : pstage;
    pstage = pstage > (CE_NUM_POINTS - 8) ? (CE_NUM_POINTS - 8) : pstage;

    // ---- async DMA: 8 contiguous embedding rows (2 KB) -> this wave's slot.
    // 32 lanes x b128 = 512 B per issue; 4 issues cover 2 KB. The instruction
    // immediate offset is added to BOTH the LDS and global addresses.
    float* slot = &lds[wave * CE_SLOT_FLOATS];
    const unsigned ldsOff =
        (unsigned)(uint64_t)(uintptr_t)slot + (unsigned)(lane * 16); // low 32b of generic LDS addr == LDS byte offset
    const uint64_t gaddr =
        (uint64_t)(uintptr_t)(emb + (size_t)pstage * CE_OUT_DIMS) + (uint64_t)(lane * 16);
    asm volatile("global_load_async_to_lds_b128 %0, %1, off"
                 :: "v"(ldsOff), "v"(gaddr) : "memory");
    asm volatile("global_load_async_to_lds_b128 %0, %1, off offset:512"
                 :: "v"(ldsOff), "v"(gaddr) : "memory");
    asm volatile("global_load_async_to_lds_b128 %0, %1, off offset:1024"
                 :: "v"(ldsOff), "v"(gaddr) : "memory");
    asm volatile("global_load_async_to_lds_b128 %0, %1, off offset:1536"
                 :: "v"(ldsOff), "v"(gaddr) : "memory");

    // ---- overlap: Hann weights while the DMA is in flight.
    // w_k = cos^2(pi*d/8) for d = xs - p, |d| < 4; the rect window is implicit
    // because cos^2 vanishes at |d| = 4. Taps outside the table are zeroed.
    float w[8];
    float wsum = 0.0f;
    #pragma unroll
    for (int k = 0; k < 8; ++k) {
        const int   p = p0 + k;
        const float d = xs - (float)p;
        const float c = __cosf(d * (float)(3.14159265358979323846 / 8.0));
        float wk = c * c;
        wk = (p >= 0 && p < CE_NUM_POINTS) ? wk : 0.0f;
        w[k] = wk;
        wsum += wk;
    }
    const float inv = 1.0f / wsum;   // normalized=True

    // ---- wait for the async copy, then accumulate from LDS.
    asm volatile("s_wait_asynccnt 0" ::: "memory");

    const int shift = p0 - pstage;               // in [-3, 4]
    const float2* slot2 = (const float2*)slot;   // lane owns dims 2*lane, 2*lane+1
    float accx = 0.0f, accy = 0.0f;
    #pragma unroll
    for (int k = 0; k < 8; ++k) {
        int row = k + shift;
        row = row < 0 ? 0 : (row > 7 ? 7 : row); // clamped rows carry w==0
        const float2 e = slot2[row * (CE_OUT_DIMS / 2) + lane];
        accx += w[k] * e.x;
        accy += w[k] * e.y;
    }

    float2 res;
    res.x = accx * inv;
    res.y = accy * inv;
    ((float2*)(out + (size_t)xi * CE_OUT_DIMS))[lane] = res; // coalesced 256 B/wave
}

extern "C" void kernel_launch(void* const* d_in, const int* in_sizes, int n_in,
                              void* d_out, int out_size, void* d_ws, size_t ws_size,
                              hipStream_t stream)
{
    (void)n_in; (void)out_size; (void)d_ws; (void)ws_size;
    const float* x   = (const float*)d_in[0];   // [128, 200] f32
    const float* emb = (const float*)d_in[1];   // [2048, 64] f32
    float* out = (float*)d_out;                 // [128, 200, 64] f32
    const int nx = in_sizes[0];                 // 25600
    const int blocks = (nx + CE_WAVES_PER_BLOCK - 1) / CE_WAVES_PER_BLOCK;
    ContinuousEmbedding_kernel<<<blocks, 256, 0, stream>>>(x, emb, out, nx);
}